// Gate_36318243455068
// MI455X (gfx1250) — compile-verified
//
#include <hip/hip_runtime.h>
#include <cstdint>

// Problem constants (from reference)
#define BATCH 16
#define LSEQ  4096
#define DCH   1024
#define TPB   256
#define SEG   16            // stage1 per-thread scan segment (TPB*SEG == LSEQ)
#define CL    256           // chunk length for the chunked scan
#define NCH   (LSEQ / CL)   // 16 chunks per batch row
#define ROWS  8             // rows per LDS tile (scalar pass)
#define NT    (CL / ROWS)
#define ROWSW 16            // rows per LDS tile (WMMA pass)
#define NTW   (CL / ROWSW)
#define EPSV  1e-20f

typedef __attribute__((ext_vector_type(4))) float v4f;
typedef __attribute__((ext_vector_type(4))) int   v4i;
typedef __attribute__((ext_vector_type(2))) float v2f;
typedef __attribute__((ext_vector_type(8))) float v8f;

// ---------------- CDNA5 async global->LDS helpers ----------------
__device__ __forceinline__ void async16(const float* g, float* l) {
#if __has_builtin(__builtin_amdgcn_global_load_async_to_lds_b128)
  __builtin_amdgcn_global_load_async_to_lds_b128(
      (__attribute__((address_space(1))) v4i*)(g),
      (__attribute__((address_space(3))) v4i*)(l),
      0, 0);
#else
#warning "CDNA5: __builtin_amdgcn_global_load_async_to_lds_b128 not available; synchronous fallback in use"
  *(v4f*)l = *(const v4f*)g;
#endif
}

#if __has_builtin(__builtin_amdgcn_s_wait_asynccnt)
#define WAIT_ASYNC(n) __builtin_amdgcn_s_wait_asynccnt(n)
#else
#warning "CDNA5: __builtin_amdgcn_s_wait_asynccnt not available; inline asm fallback in use"
#define WAIT_ASYNC(n) asm volatile("s_wait_asynccnt %0" ::"n"(n))
#endif

// ---------------- CDNA5 WMMA f32 16x16x4 ----------------
__device__ __forceinline__ v8f wmma4_f32(v2f a, v2f b, v8f c) {
#if __has_builtin(__builtin_amdgcn_wmma_f32_16x16x4_f32)
  return __builtin_amdgcn_wmma_f32_16x16x4_f32(false, a, false, b, (short)0, c,
                                               false, false);
#else
#warning "CDNA5: __builtin_amdgcn_wmma_f32_16x16x4_f32 not available"
  return c;
#endif
}

__device__ __forceinline__ float sigm(float t) { return 1.0f / (1.0f + __expf(-t)); }

// ---------------- Pass 1: a, w, z-scan (scalar per (b,l)), chunk products ----
__global__ __launch_bounds__(TPB) void gate_stage1(
    const float* __restrict__ x, const float* __restrict__ u,
    float* __restrict__ a_g, float* __restrict__ w_g,
    float* __restrict__ zinv_g, float* __restrict__ achunk_g) {
  __shared__ float sa[LSEQ];
  __shared__ float sw[LSEQ];
  __shared__ float sz[LSEQ];
  __shared__ float pA[TPB];
  __shared__ float pB[TPB];

  const int tid = threadIdx.x;
  const int b   = blockIdx.x;

  for (int l = tid; l < LSEQ; l += TPB) {
    const size_t row = (size_t)(b * LSEQ + l);
    float x0 = x[row * DCH + 0];
    float x1 = x[row * DCH + 1];
    float un = u[b * LSEQ + l];
    float g  = -__logf(-__logf(un + EPSV) + EPSV);
    float av = sigm((x0 - 1.0f + g) * 2.0f);   // C=1.0, 1/T = 2
    float wv = sigm(x1);
    if (l == 0) av = 0.0f;  // folds z0=w0, s0=x0*w0 into the uniform recurrence
    sa[l] = av; sw[l] = wv;
    a_g[row] = av; w_g[row] = wv;
  }
  __syncthreads();

  {
    const int s0 = tid * SEG;
    float A = 1.0f, Bv = 0.0f;
#pragma unroll
    for (int i = 0; i < SEG; ++i) {
      float av = sa[s0 + i], wv = sw[s0 + i];
      A  = av * A;
      Bv = fmaf(av, Bv, wv);
    }
    pA[tid] = A; pB[tid] = Bv;
  }
  __syncthreads();

  if (tid == 0) {
    float carry = 0.0f;
#pragma unroll 4
    for (int t = 0; t < TPB; ++t) {
      float nxt = fmaf(pA[t], carry, pB[t]);
      pB[t] = carry;       // exclusive init for segment t
      carry = nxt;
    }
  }
  __syncthreads();

  {
    const int s0 = tid * SEG;
    float z = pB[tid];
#pragma unroll
    for (int i = 0; i < SEG; ++i) {
      z = fmaf(sa[s0 + i], z, sw[s0 + i]);
      sz[s0 + i] = z;
    }
  }
  __syncthreads();

  for (int l = tid; l < LSEQ; l += TPB)
    zinv_g[(size_t)(b * LSEQ + l)] = 1.0f / sz[l];

  if (tid < NCH) {
    float P = 1.0f;
#pragma unroll
    for (int t = tid * (CL / SEG); t < (tid + 1) * (CL / SEG); ++t) P *= pA[t];
    achunk_g[b * NCH + tid] = P;
  }
}

// ---------------- Pass 2: chunk-local s-scan (zero init), carries out -------
__global__ __launch_bounds__(TPB) void gate_chunk_local(
    const float* __restrict__ x, const float* __restrict__ a_g,
    const float* __restrict__ w_g, float* __restrict__ slocal_g) {
  __shared__ float s_a[CL];
  __shared__ float s_w[CL];
  __shared__ float tile[2][ROWS * TPB];

  const int tid = threadIdx.x;
  const int b = blockIdx.z, c = blockIdx.y, d0 = blockIdx.x * TPB;
  const size_t rowbase = (size_t)(b * LSEQ + c * CL);

  s_a[tid] = a_g[rowbase + tid];   // CL == TPB
  s_w[tid] = w_g[rowbase + tid];

  const int row0 = tid >> 6;            // 0..3
  const int col  = (tid & 63) * 4;
  const int f0 = tid * 4, f1 = ROWS * TPB / 2 + tid * 4;
  const float* g0 = x + (rowbase + row0) * DCH + d0 + col;       // rows 0..3
  const float* g1 = x + (rowbase + 4 + row0) * DCH + d0 + col;   // rows 4..7

  async16(g0, &tile[0][f0]);
  async16(g1, &tile[0][f1]);

  float s = 0.0f;
  for (int t = 0; t < NT; ++t) {
    const int cb = t & 1, nb = (t + 1) & 1;
    if (t + 1 < NT) {
      const size_t adv = (size_t)(t + 1) * ROWS * DCH;
      async16(g0 + adv, &tile[nb][f0]);
      async16(g1 + adv, &tile[nb][f1]);
      WAIT_ASYNC(2);
    } else {
      WAIT_ASYNC(0);
    }
    __syncthreads();
#pragma unroll
    for (int r = 0; r < ROWS; ++r) {
      float xv = tile[cb][r * TPB + tid];
      const int l = t * ROWS + r;
      s = fmaf(s_a[l], s, xv * s_w[l]);
    }
    __syncthreads();
  }
  slocal_g[((size_t)(b * NCH + c)) * DCH + d0 + tid] = s;
}

// ---------------- Pass 3: sequential combine of chunk carries ---------------
__global__ __launch_bounds__(TPB) void gate_combine(
    const float* __restrict__ achunk_g, const float* __restrict__ slocal_g,
    float* __restrict__ sinit_g) {
  const int d = blockIdx.x * TPB + threadIdx.x;
  const int b = blockIdx.y;
  float s = 0.0f;
#pragma unroll
  for (int c = 0; c < NCH; ++c) {
    const size_t idx = ((size_t)(b * NCH + c)) * DCH + d;
    sinit_g[idx] = s;                               // exclusive init for chunk c
    s = fmaf(achunk_g[b * NCH + c], s, slocal_g[idx]);
  }
}

// ---------------- Pass 4: WMMA chunked scan, write y ------------------------
// Per 16-row block t:  S = T16 (16x16 lower-tri) x U (16x16 cols) + p16 (x) carry
// A-frag (16x4 f32): lanes 0-15 M=0..15; VGPR0 K=2*half, VGPR1 K=2*half+1
// B-frag (4x16 f32): lanes 0-15 N; VGPR0 K=2*half, VGPR1 K=2*half+1
// C/D (16x16 f32):   lanes 0-15 N; VGPR r holds M = r + 8*half
__global__ __launch_bounds__(TPB) void gate_apply(
    const float* __restrict__ x, const float* __restrict__ a_g,
    const float* __restrict__ w_g, const float* __restrict__ zinv_g,
    const float* __restrict__ sinit_g, float* __restrict__ y) {
  __shared__ float s_a[CL];
  __shared__ float s_w[CL];
  __shared__ float s_z[CL];
  __shared__ float sT[16 * 17];     // padded stride to dodge bank conflicts
  __shared__ float p16[16];
  __shared__ float tile[2][ROWSW * TPB];

  const int tid = threadIdx.x;
  const int b = blockIdx.z, c = blockIdx.y, d0 = blockIdx.x * TPB;
  const size_t rowbase = (size_t)(b * LSEQ + c * CL);

  s_a[tid] = a_g[rowbase + tid];
  s_w[tid] = w_g[rowbase + tid];
  s_z[tid] = zinv_g[rowbase + tid];

  const int lane = tid & 31;
  const int wv   = tid >> 5;        // wave index, wave owns 32 columns
  const int n0   = wv * 32;
  const int nn   = lane & 15;
  const int half = lane >> 4;       // 0/1
  const int m16  = lane & 15;       // A-matrix M index

  // async staging: 16-row x 256-col tile; 4 x 16B per thread per tile
  const int rj   = tid >> 6;                    // 0..3
  const int colg = (tid & 63) * 4;
  const float* gb = x + rowbase * DCH + d0 + colg;
  const int fj = tid * 4;

#pragma unroll
  for (int j = 0; j < 4; ++j)
    async16(gb + (size_t)(j * 4 + rj) * DCH, &tile[0][j * 1024 + fj]);

  float carryA = sinit_g[((size_t)(b * NCH + c)) * DCH + d0 + n0 + nn];
  float carryB = sinit_g[((size_t)(b * NCH + c)) * DCH + d0 + n0 + 16 + nn];

  for (int t = 0; t < NTW; ++t) {
    const int cb = t & 1, nb = (t + 1) & 1;
    if (t + 1 < NTW) {
      const size_t adv = (size_t)(t + 1) * ROWSW * DCH;
#pragma unroll
      for (int j = 0; j < 4; ++j)
        async16(gb + adv + (size_t)(j * 4 + rj) * DCH, &tile[nb][j * 1024 + fj]);
      WAIT_ASYNC(4);
    } else {
      WAIT_ASYNC(0);
    }
    __syncthreads();                // tile t resident; previous sT consumed

    // build T16 / p16 for this 16-row block: one element per thread
    {
      const int mm = tid >> 4, kk = tid & 15;
      float v;
      if (mm < kk) {
        v = 0.0f;
      } else {
        v = 1.0f;
        for (int i = kk + 1; i <= mm; ++i) v *= s_a[t * 16 + i];
      }
      sT[mm * 17 + kk] = v;
      if (kk == 0) p16[mm] = v * s_a[t * 16];
    }
    __syncthreads();

    const float* tx = tile[cb];
    v8f cA, cB;
#pragma unroll
    for (int r = 0; r < 8; ++r) {
      const int m = r + 8 * half;
      cA[r] = p16[m] * carryA;
      cB[r] = p16[m] * carryB;
    }

#pragma unroll
    for (int kk = 0; kk < 4; ++kk) {
      const int k0 = kk * 4 + 2 * half;
      v2f a, bA, bB;
      a[0] = sT[m16 * 17 + k0];
      a[1] = sT[m16 * 17 + k0 + 1];
      const float w0 = s_w[t * 16 + k0];
      const float w1 = s_w[t * 16 + k0 + 1];
      bA[0] = tx[k0 * TPB + n0 + nn] * w0;
      bA[1] = tx[(k0 + 1) * TPB + n0 + nn] * w1;
      bB[0] = tx[k0 * TPB + n0 + 16 + nn] * w0;
      bB[1] = tx[(k0 + 1) * TPB + n0 + 16 + nn] * w1;
      cA = wmma4_f32(a, bA, cA);
      cB = wmma4_f32(a, bB, cB);
    }

    // next block's carry = S[15,:] (C VGPR 7, lanes 16..31)
    carryA = __shfl(cA[7], 16 + nn, 32);
    carryB = __shfl(cB[7], 16 + nn, 32);

#pragma unroll
    for (int r = 0; r < 8; ++r) {
      const int m = r + 8 * half;
      const int l = t * 16 + m;
      const float zi = s_z[l];
      float oA = cA[r] * zi;
      float oB = cB[r] * zi;
      if (c == 0 && l == 0) {       // y[:,0] = x[:,0] (raw)
        oA = tx[n0 + nn];
        oB = tx[n0 + 16 + nn];
      }
      y[(rowbase + l) * DCH + d0 + n0 + nn] = oA;
      y[(rowbase + l) * DCH + d0 + n0 + 16 + nn] = oB;
    }
    __syncthreads();                // done with tile t / sT before overwrite
  }
}

// ---------------- x_pad pass-through (second tuple output) ------------------
__global__ __launch_bounds__(TPB) void gate_pad(
    const unsigned char* __restrict__ pad, float* __restrict__ outp, int n) {
  const int i = blockIdx.x * TPB + threadIdx.x;
  if (i < n) outp[i] = pad[i] ? 1.0f : 0.0f;
}

// ---------------- host side -------------------------------------------------
extern "C" void kernel_launch(void* const* d_in, const int* in_sizes, int n_in,
                              void* d_out, int out_size, void* d_ws, size_t ws_size,
                              hipStream_t stream) {
  const float* x   = (const float*)d_in[0];
  const unsigned char* pad = (const unsigned char*)d_in[1];
  const float* u   = (const float*)d_in[2];
  float* y = (float*)d_out;

  float* ws = (float*)d_ws;
  float* a_ws      = ws;                         // B*L        = 65536
  float* w_ws      = a_ws   + BATCH * LSEQ;      // B*L
  float* zinv_ws   = w_ws   + BATCH * LSEQ;      // B*L
  float* achunk_ws = zinv_ws + BATCH * LSEQ;     // B*NCH      = 256
  float* slocal_ws = achunk_ws + BATCH * NCH;    // B*NCH*D    = 262144
  float* sinit_ws  = slocal_ws + (size_t)BATCH * NCH * DCH;

  gate_stage1<<<BATCH, TPB, 0, stream>>>(x, u, a_ws, w_ws, zinv_ws, achunk_ws);
  gate_chunk_local<<<dim3(DCH / TPB, NCH, BATCH), TPB, 0, stream>>>(
      x, a_ws, w_ws, slocal_ws);
  gate_combine<<<dim3(DCH / TPB, BATCH), TPB, 0, stream>>>(
      achunk_ws, slocal_ws, sinit_ws);
  gate_apply<<<dim3(DCH / TPB, NCH, BATCH), TPB, 0, stream>>>(
      x, a_ws, w_ws, zinv_ws, sinit_ws, y);
  gate_pad<<<(BATCH * LSEQ) / TPB, TPB, 0, stream>>>(
      pad, y + (size_t)BATCH * LSEQ * DCH, BATCH * LSEQ);
}